// SIGReg_35553739276595
// MI455X (gfx1250) — compile-verified
//
#include <hip/hip_runtime.h>
#include <math.h>

// Problem constants (match reference setup_inputs)
#define BATCH 8
#define NROWS 1024
#define CDIM  1024
#define VDIM  256
#define KNOTS 17
#define DT    0.1875f            // 3.0 / (KNOTS-1)

typedef __attribute__((ext_vector_type(2))) float v2f;
typedef __attribute__((ext_vector_type(8))) float v8f;

#define LDS_STRIDE 260           // 256 + 4 pad -> stride mod 64 == 4 (conflict-free b64 reads)
#define KCHUNK     256

#if defined(__has_builtin)
#if __has_builtin(__builtin_amdgcn_global_load_async_to_lds_b128)
#define HAVE_ASYNC_LDS 1
#endif
#endif

#ifdef HAVE_ASYNC_LDS
typedef int v4i_t __attribute__((vector_size(16)));
typedef __attribute__((address_space(1))) v4i_t* gv4_p;   // global (AS1)
typedef __attribute__((address_space(3))) v4i_t* lv4_p;   // LDS (AS3)

__device__ __forceinline__ void async_copy_b128(const void* gsrc, void* ldst) {
    // GLOBAL_LOAD_ASYNC_TO_LDS_B128: memory -> LDS DMA, tracked by ASYNCcnt.
    __builtin_amdgcn_global_load_async_to_lds_b128(
        (gv4_p)const_cast<void*>(gsrc), (lv4_p)ldst, 0, 0);
}

__device__ __forceinline__ void wait_asynccnt0() {
#if __has_builtin(__builtin_amdgcn_s_wait_asynccnt)
    __builtin_amdgcn_s_wait_asynccnt(0);
#else
    asm volatile("s_wait_asynccnt 0x0" ::: "memory");
#endif
}
#endif

// ---------------------------------------------------------------------------
// Phase 1: fused GEMM (f32 WMMA) + characteristic-function accumulation.
// grid = BATCH * (NROWS/16) blocks, block = 512 threads (16 waves).
// Each block owns one (b, 16-row N-tile); wave w owns V columns [16w, 16w+16).
// proj (32 MB) is read from HBM exactly once; A (1 MB) stays L2-resident.
// ---------------------------------------------------------------------------
__global__ __launch_bounds__(512)
void sigreg_gemm_cf_kernel(const float* __restrict__ proj,
                           const float* __restrict__ Amat,
                           float* __restrict__ cosAcc,   // [B][V][KNOTS]
                           float* __restrict__ sinAcc)   // [B][V][KNOTS]
{
    __shared__ float ldsProj[16 * LDS_STRIDE];           // 16 rows x KCHUNK cols (padded)

    const int bidx  = blockIdx.x >> 6;                   // batch
    const int ntile = blockIdx.x & 63;                   // N tile
    const int row0  = ntile * 16;

    const int lane = threadIdx.x & 31;
    const int wave = threadIdx.x >> 5;

    // WMMA f32 16x16x4 fragment addressing (wave32):
    //   A-frag: lane l holds A[m][kb], A[m][kb+1], m = l&15, kb = 2*(l>>4)
    //   B-frag: lane l holds B[kb][n], B[kb+1][n],  n = l&15
    //   C/D   : VGPR r -> row (r + 8*(l>>4)), col l&15
    const int m_l  = lane & 15;
    const int kb   = (lane >> 4) * 2;
    const int ncol = wave * 16 + (lane & 15);            // global V column

    v8f acc = {};

    const float* projTile = proj + ((size_t)(bidx * NROWS + row0)) * CDIM;

    for (int k0 = 0; k0 < CDIM; k0 += KCHUNK) {
        __syncthreads();                                 // LDS reuse guard
        // Stage 16 x KCHUNK chunk of proj into LDS.
        // 16*64 = 1024 float4s, 512 threads -> 2 each, fully coalesced.
        for (int i = threadIdx.x; i < 16 * (KCHUNK / 4); i += 512) {
            const int m  = i >> 6;
            const int c4 = i & 63;
            const float* gsrc = projTile + (size_t)m * CDIM + k0 + c4 * 4;
            float*       ldst = &ldsProj[m * LDS_STRIDE + c4 * 4];
#ifdef HAVE_ASYNC_LDS
            async_copy_b128(gsrc, ldst);                 // memory -> LDS, no VGPR bounce
#else
            *(float4*)ldst = *(const float4*)gsrc;
#endif
        }
#ifdef HAVE_ASYNC_LDS
        wait_asynccnt0();
#endif
        __syncthreads();

        #pragma unroll 8
        for (int kk = 0; kk < KCHUNK; kk += 4) {
            v2f a, b;
            a.x = ldsProj[m_l * LDS_STRIDE + kk + kb];
            a.y = ldsProj[m_l * LDS_STRIDE + kk + kb + 1];
            const int krow = k0 + kk + kb;
            b.x = Amat[(size_t)krow * VDIM + ncol];
            b.y = Amat[(size_t)(krow + 1) * VDIM + ncol];
            acc = __builtin_amdgcn_wmma_f32_16x16x4_f32(
                false, a, false, b, (short)0, acc, false, false);
        }
    }

    // --- CF accumulation via angle-addition recurrence ---------------------
    // cos(x*t_k), sin(x*t_k) for t_k = k*DT: 2 trans ops + 16 rotations per x.
    float csum[KNOTS], ssum[KNOTS];
    #pragma unroll
    for (int k = 0; k < KNOTS; ++k) { csum[k] = 0.0f; ssum[k] = 0.0f; }

    #pragma unroll
    for (int r = 0; r < 8; ++r) {
        const float x  = acc[r];
        const float th = x * DT;
        const float c1 = __cosf(th);
        const float s1 = __sinf(th);
        float c = 1.0f, s = 0.0f;                        // k = 0: cos=1, sin=0
        #pragma unroll
        for (int k = 0; k < KNOTS; ++k) {
            csum[k] += c;
            ssum[k] += s;
            const float cn = __builtin_fmaf(c, c1, -s * s1);
            const float sn = __builtin_fmaf(c, s1,  s * c1);
            c = cn; s = sn;
        }
    }

    // Fold the two row-halves (lane l and l+16 hold the same V column),
    // then lanes 0-15 commit per-(b,v,k) partial sums with native fp32 atomics.
    const size_t base = ((size_t)bidx * VDIM + ncol) * KNOTS;
    #pragma unroll
    for (int k = 0; k < KNOTS; ++k) {
        const float ct = csum[k] + __shfl_xor(csum[k], 16, 32);
        const float st = ssum[k] + __shfl_xor(ssum[k], 16, 32);
        if (lane < 16) {
            unsafeAtomicAdd(&cosAcc[base + k], ct);
            unsafeAtomicAdd(&sinAcc[base + k], st);
        }
    }
}

// ---------------------------------------------------------------------------
// Phase 2: (B,V,17) accumulators -> scalar statistic mean.
// 8 blocks x 256 threads: one (b,v) pair per thread, block-reduce, then one
// fp32 atomic per block into out[0] (zeroed by hipMemsetAsync each call).
// ---------------------------------------------------------------------------
__global__ __launch_bounds__(256)
void sigreg_reduce_kernel(const float* __restrict__ ws, float* __restrict__ out)
{
    const float* cosAcc = ws;
    const float* sinAcc = ws + (size_t)BATCH * VDIM * KNOTS;

    const int i = blockIdx.x * 256 + threadIdx.x;        // (b,v) pair, 0..2047
    const float* cp = cosAcc + (size_t)i * KNOTS;
    const float* sp = sinAcc + (size_t)i * KNOTS;

    float stat = 0.0f;
    #pragma unroll
    for (int k = 0; k < KNOTS; ++k) {
        const float t   = DT * (float)k;
        const float phi = __expf(-0.5f * t * t);
        const float w   = ((k == 0) || (k == KNOTS - 1)) ? DT * phi : 2.0f * DT * phi;
        const float cm  = cp[k] * (1.0f / (float)NROWS);
        const float sm  = sp[k] * (1.0f / (float)NROWS);
        const float err = (cm - phi) * (cm - phi) + sm * sm;
        stat = __builtin_fmaf(err, w, stat);
    }
    float local = stat * (float)NROWS;                   // statistic = (err @ w) * N

    __shared__ float red[256];
    red[threadIdx.x] = local;
    __syncthreads();
    for (int off = 128; off > 0; off >>= 1) {
        if (threadIdx.x < off) red[threadIdx.x] += red[threadIdx.x + off];
        __syncthreads();
    }
    if (threadIdx.x == 0)
        unsafeAtomicAdd(out, red[0] * (1.0f / (float)(BATCH * VDIM)));
}

// ---------------------------------------------------------------------------
extern "C" void kernel_launch(void* const* d_in, const int* in_sizes, int n_in,
                              void* d_out, int out_size, void* d_ws, size_t ws_size,
                              hipStream_t stream)
{
    (void)in_sizes; (void)n_in; (void)out_size; (void)ws_size;

    const float* proj = (const float*)d_in[0];           // (B, N, C) f32
    const float* Amat = (const float*)d_in[1];           // (C, V) f32
    float*       out  = (float*)d_out;                   // scalar f32
    float*       ws   = (float*)d_ws;

    const size_t accElems = 2ull * BATCH * VDIM * KNOTS; // cosAcc + sinAcc
    (void)hipMemsetAsync(d_ws, 0, accElems * sizeof(float), stream);
    (void)hipMemsetAsync(d_out, 0, sizeof(float), stream); // out accumulated atomically

    float* cosAcc = ws;
    float* sinAcc = ws + (size_t)BATCH * VDIM * KNOTS;

    sigreg_gemm_cf_kernel<<<BATCH * (NROWS / 16), 512, 0, stream>>>(proj, Amat, cosAcc, sinAcc);
    sigreg_reduce_kernel<<<(BATCH * VDIM) / 256, 256, 0, stream>>>(ws, out);
}